// Model_75204877353282
// MI455X (gfx1250) — compile-verified
//
#include <hip/hip_runtime.h>
#include <math.h>

// Problem sizes (fixed by the reference).
#define BB 64
#define LL 512
#define SS 512
#define HH 768
#define CC 9
#define NPAD 16   // N padded to 16 for the 16x16 WMMA tile

typedef float v2f __attribute__((ext_vector_type(2)));
typedef float v8f __attribute__((ext_vector_type(8)));

// ---------------------------------------------------------------------------
// Kernel 0: zero the scratch accumulators (harness does not re-zero d_ws).
// ---------------------------------------------------------------------------
__global__ void zero_ws_kernel(float* __restrict__ wsums, float* __restrict__ loss_acc) {
    int i = blockIdx.x * blockDim.x + threadIdx.x;
    if (i < BB * SS * CC) wsums[i] = 0.0f;
    if (i < 2) loss_acc[i] = 0.0f;
}

// ---------------------------------------------------------------------------
// Kernel 1: per-batch subword->word map.
//   sidx[l] = cumsum(attention_mask)-1 ; ends[s] = cumsum(ids_lens)
//   word_of[b,l] = searchsorted_right(ends, sidx[l])  (or -1 if masked)
// One block of 512 threads per batch element; scans in LDS.
// ---------------------------------------------------------------------------
__global__ __launch_bounds__(512) void build_word_map_kernel(
    const int* __restrict__ am, const int* __restrict__ ids_lens,
    int* __restrict__ word_of) {
    __shared__ int ends[SS];
    __shared__ int sidx[LL];
    const int b = blockIdx.x;
    const int t = threadIdx.x;

    ends[t] = ids_lens[b * SS + t];
    const int a = am[b * LL + t];
    sidx[t] = a;
    __syncthreads();

    // Hillis-Steele inclusive scans of both arrays.
    for (int off = 1; off < 512; off <<= 1) {
        int ve = (t >= off) ? ends[t - off] : 0;
        int vs = (t >= off) ? sidx[t - off] : 0;
        __syncthreads();
        ends[t] += ve;
        sidx[t] += vs;
        __syncthreads();
    }

    int w = -1;
    if (a) {
        const int s = sidx[t] - 1;             // position within valid region
        int lo = 0, hi = SS;                   // first idx with ends[idx] > s
        while (lo < hi) {
            int mid = (lo + hi) >> 1;
            if (ends[mid] <= s) lo = mid + 1; else hi = mid;
        }
        w = lo;
    }
    word_of[b * LL + t] = w;
}

// ---------------------------------------------------------------------------
// Kernel 2: fused projection + segment pooling.
// Tall-skinny GEMM [B*L, H] @ [H, 16(pad of 9)] with V_WMMA_F32_16X16X4_F32,
// then atomic segment-sum of the 9 logits per subword into its word bucket.
// One wave per 16-row tile; 8 waves (256 threads) per block; W staged in LDS.
// ---------------------------------------------------------------------------
__global__ __launch_bounds__(256) void proj_pool_kernel(
    const float* __restrict__ X,        // [B*L, H] bert_out
    const float* __restrict__ W,        // [H, C]
    const int*   __restrict__ word_of,  // [B*L]
    float*       __restrict__ wsums) {  // [B*S, C]
    __shared__ float wlds[HH * NPAD];   // W zero-padded to [H,16] = 48 KB
    const int t = threadIdx.x;
    for (int i = t; i < HH * NPAD; i += 256) {
        const int kk = i >> 4, nn = i & 15;
        wlds[i] = (nn < CC) ? W[kk * CC + nn] : 0.0f;
    }
    __syncthreads();

    const int wave = t >> 5;
    const int lane = t & 31;
    const int tile = blockIdx.x * 8 + wave;   // 2048 tiles total, exact cover
    const int r0   = tile << 4;               // first of 16 rows (subwords)

    const int lane_lo = lane & 15;
    const int khalf   = (lane >> 4) << 1;     // lanes 0-15 hold K=0,1 ; 16-31 K=2,3
    const float* rowp = X + (size_t)(r0 + lane_lo) * HH + khalf;
    const float* blds = wlds + khalf * NPAD + lane_lo;

    v8f acc = {};
    for (int k0 = 0; k0 < HH; k0 += 4) {
        if ((k0 & 63) == 0)                    // uniform branch; EXEC unchanged
            __builtin_prefetch(rowp + k0 + 256, 0, 0);   // global_prefetch_b8
        // A tile (16x4 f32): contiguous K pair per lane -> global_load_b64.
        v2f a = *(const v2f*)(rowp + k0);
        // B tile (4x16 f32) from LDS: rows khalf+k0, khalf+k0+1 at column lane_lo.
        v2f bm;
        bm.x = blds[(k0 + 0) * NPAD];
        bm.y = blds[(k0 + 1) * NPAD];
        acc = __builtin_amdgcn_wmma_f32_16x16x4_f32(
            /*neg_a=*/false, a, /*neg_b=*/false, bm,
            /*c_mod=*/(short)0, acc, /*reuse_a=*/false, /*reuse_b=*/false);
    }

    // D layout: VGPR i -> row M = i (lanes 0-15) or 8+i (lanes 16-31), N = lane&15.
    const int n = lane_lo;
    if (n < CC) {
        const int mbase = r0 + ((lane >> 4) << 3);
        #pragma unroll
        for (int i = 0; i < 8; ++i) {
            const int row = mbase + i;            // flat subword index b*L + l
            const int w = word_of[row];
            if (w >= 0) {
                const int b = row >> 9;           // / LL
                atomicAdd(&wsums[((size_t)b * SS + w) * CC + n], acc[i]);
            }
        }
    }
}

// ---------------------------------------------------------------------------
// Kernel 3: per-word finalize: mean, +bias, log-softmax(9), NLL, argmax.
// One thread per (b, s); block-reduced loss accumulation.
// ---------------------------------------------------------------------------
__global__ __launch_bounds__(256) void finalize_kernel(
    const float* __restrict__ wsums, const int* __restrict__ ids_lens,
    const float* __restrict__ bias,  const int* __restrict__ label_ids,
    float* __restrict__ pred_out, float* __restrict__ loss_acc) {
    const int idx = blockIdx.x * blockDim.x + threadIdx.x;   // [0, B*S)
    const int cnt = ids_lens[idx];
    const float inv = (cnt > 0) ? (1.0f / (float)cnt) : 0.0f;

    float lg[CC];
    float mx = -INFINITY;
    int amx = 0;
    #pragma unroll
    for (int c = 0; c < CC; ++c) {
        const float v = wsums[(size_t)idx * CC + c] * inv + bias[c];
        lg[c] = v;
        if (v > mx) { mx = v; amx = c; }   // first max (strict >), matches jnp.argmax
    }
    float se = 0.0f;
    #pragma unroll
    for (int c = 0; c < CC; ++c) se += expf(lg[c] - mx);
    const float lse = logf(se);

    int lab = label_ids[idx];
    lab = lab < 0 ? 0 : (lab > CC - 1 ? CC - 1 : lab);
    const float valid = (cnt > 0) ? 1.0f : 0.0f;   // label_mask == ids_lens > 0
    const float nll = -(lg[lab] - mx - lse);

    pred_out[idx] = (float)amx;

    __shared__ float rn[256];
    __shared__ float rd[256];
    const int t = threadIdx.x;
    rn[t] = nll * valid;
    rd[t] = valid;
    __syncthreads();
    for (int off = 128; off > 0; off >>= 1) {
        if (t < off) { rn[t] += rn[t + off]; rd[t] += rd[t + off]; }
        __syncthreads();
    }
    if (t == 0) {
        atomicAdd(&loss_acc[0], rn[0]);
        atomicAdd(&loss_acc[1], rd[0]);
    }
}

// ---------------------------------------------------------------------------
// Kernel 4: loss = num / max(den, 1)
// ---------------------------------------------------------------------------
__global__ void loss_final_kernel(const float* __restrict__ loss_acc,
                                  float* __restrict__ out0) {
    if (threadIdx.x == 0 && blockIdx.x == 0)
        out0[0] = loss_acc[0] / fmaxf(loss_acc[1], 1.0f);
}

// ---------------------------------------------------------------------------
extern "C" void kernel_launch(void* const* d_in, const int* in_sizes, int n_in,
                              void* d_out, int out_size, void* d_ws, size_t ws_size,
                              hipStream_t stream) {
    const float* X        = (const float*)d_in[0];  // bert_out [B,L,H]
    const float* W        = (const float*)d_in[1];  // [H,C]
    const float* bias     = (const float*)d_in[2];  // [C]
    const int*   am       = (const int*)d_in[3];    // attention_mask [B,L]
    const int*   ids_lens = (const int*)d_in[4];    // [B,S]
    const int*   label_ids= (const int*)d_in[5];    // [B,S]
    // d_in[6] (label_mask) intentionally unused: identical to ids_lens > 0.

    float* out = (float*)d_out;                     // [0]=loss, [1..B*S]=pred

    char* ws = (char*)d_ws;
    float* wsums   = (float*)ws;                                        // B*S*C f32
    int*   word_of = (int*)(ws + sizeof(float) * BB * SS * CC);         // B*L i32
    float* loss_acc= (float*)(ws + sizeof(float) * BB * SS * CC
                                 + sizeof(int)   * BB * LL);            // 2 f32

    zero_ws_kernel<<<(BB * SS * CC + 255) / 256, 256, 0, stream>>>(wsums, loss_acc);
    build_word_map_kernel<<<BB, 512, 0, stream>>>(am, ids_lens, word_of);
    proj_pool_kernel<<<(BB * LL / 16) / 8, 256, 0, stream>>>(X, W, word_of, wsums);
    finalize_kernel<<<(BB * SS) / 256, 256, 0, stream>>>(wsums, ids_lens, bias,
                                                         label_ids, out + 1, loss_acc);
    loss_final_kernel<<<1, 1, 0, stream>>>(loss_acc, out);
}